// UniSAGEConv_48550310314283
// MI455X (gfx1250) — compile-verified
//
#include <hip/hip_runtime.h>

typedef __attribute__((ext_vector_type(2))) float v2f;
typedef __attribute__((ext_vector_type(8))) float v8f;

#define D 128
// Packed-B LDS row stride (float2 columns per K-pair row). 144 => 2*144 dwords = 288
// = 32 (mod 64) banks, so lanes 0-15 (kp) and 16-31 (kp+1) hit disjoint bank
// halves on the ds_load_b64 => conflict-free 32-lane B-fragment fetch.
#define PDIM 144
// LDS: 64 K-pairs * PDIM cols * 2 floats = 18432 floats = 72 KB
#define SWSZ (64 * PDIM * 2)

// Pack W[k][n] (k:0..127, n:0..127) so that (W[k][n], W[k+1][n]) for even k
// are adjacent: sW[((k>>1)*PDIM + n)*2 + (k&1)]
__device__ __forceinline__ void pack_w(float* sW, const float* __restrict__ W,
                                       int tid) {
  for (int i = tid; i < D * D; i += 256) {
    const int k = i >> 7;
    const int n = i & (D - 1);
    sW[((((k >> 1) * PDIM) + n) << 1) | (k & 1)] = W[i];
  }
}

// ---------------------------------------------------------------------------
// Zero-fill scratch (accumulation tables must be re-zeroed every call).
// ---------------------------------------------------------------------------
__global__ void hg_zero(float* __restrict__ p, long n) {
  long i = (long)blockIdx.x * blockDim.x + threadIdx.x;
  long stride = (long)gridDim.x * blockDim.x;
  for (; i < n; i += stride) p[i] = 0.0f;
}

// ---------------------------------------------------------------------------
// S[rows x 128] = (A * rowscale) @ W[128 x 128], fp32 WMMA 16x16x4.
// If cnt != nullptr, rowscale = 1/max(cnt[row],1) (fused mean divide).
// One wave per 16-row strip; 8 accumulator tiles cover all 128 output cols.
// W pair-packed in LDS so each B fragment is one aligned ds_load_b64.
// ---------------------------------------------------------------------------
__global__ __launch_bounds__(256) void hg_gemm128(
    const float* __restrict__ A, const float* __restrict__ cnt,
    const float* __restrict__ W, float* __restrict__ S, int rows) {
  __shared__ float sW[SWSZ];
  const int tid = threadIdx.x;
  pack_w(sW, W, tid);
  __syncthreads();

  const int wave = tid >> 5;
  const int lane = tid & 31;
  const int half = lane >> 4;   // which K-pair of the 4-wide step this lane holds
  const int lrow = lane & 15;
  const int m0 = blockIdx.x * 128 + wave * 16;
  int r = m0 + lrow;
  int rc = (r < rows) ? r : (rows - 1);  // clamp loads; EXEC stays full for WMMA

  float scale = 1.0f;
  if (cnt) {
    float c = cnt[rc];
    scale = 1.0f / (c > 1.0f ? c : 1.0f);
  }
  const float* arow = A + (long)rc * D;

  v8f acc[8] = {};

  for (int k0 = 0; k0 < D; k0 += 4) {
    const int kA = k0 + 2 * half;
    v2f a;
    a.x = arow[kA] * scale;
    a.y = arow[kA + 1] * scale;
    const int kp = (k0 >> 1) + half;  // packed K-pair row
#pragma unroll
    for (int t = 0; t < 8; ++t) {
      const int n = t * 16 + lrow;
      const v2f b = *(const v2f*)&sW[(kp * PDIM + n) << 1];
      acc[t] = __builtin_amdgcn_wmma_f32_16x16x4_f32(
          false, a, false, b, (short)0, acc[t], false, false);
    }
  }

#pragma unroll
  for (int t = 0; t < 8; ++t) {
    const int n = t * 16 + lrow;
#pragma unroll
    for (int rr = 0; rr < 8; ++rr) {
      const int m = m0 + rr + 8 * half;
      if (m < rows) S[(long)m * D + n] = acc[t][rr];
    }
  }
}

// ---------------------------------------------------------------------------
// Scatter-sum: one wave per incidence; lane moves a float4 of features.
// dstSum[dstIdx[e]] += src[srcIdx[e]];  dstCnt[dstIdx[e]] += 1.
// Tables live in L2 (<= 51 MB << 192 MB), so atomics resolve at L2 rate.
// ---------------------------------------------------------------------------
__global__ __launch_bounds__(256) void hg_scatter(
    const float* __restrict__ src, const int* __restrict__ srcIdx,
    const int* __restrict__ dstIdx, float* __restrict__ dstSum,
    float* __restrict__ dstCnt, int nE) {
  const int e = blockIdx.x * 8 + (threadIdx.x >> 5);
  const int lane = threadIdx.x & 31;
  if (e >= nE) return;
  const int s = srcIdx[e];
  const int d = dstIdx[e];
  const float4* sp = (const float4*)(src + (long)s * D);
  float4 v = sp[lane];
  float* dp = dstSum + (long)d * D + lane * 4;
  atomicAdd(dp + 0, v.x);
  atomicAdd(dp + 1, v.y);
  atomicAdd(dp + 2, v.z);
  atomicAdd(dp + 3, v.w);
  if (lane == 0) atomicAdd(dstCnt + d, 1.0f);
}

// ---------------------------------------------------------------------------
// out = relu( xs @ Wu[0:128] + (nsum/max(ncnt,1)) @ Wu[128:256] + b )
// Two K=128 WMMA passes sharing one packed LDS weight panel; bias preloaded
// into the accumulators; ReLU fused at store.
// ---------------------------------------------------------------------------
__global__ __launch_bounds__(256) void hg_final(
    const float* __restrict__ xs, const float* __restrict__ nsum,
    const float* __restrict__ ncnt, const float* __restrict__ Wu,
    const float* __restrict__ bias, float* __restrict__ out, int rows) {
  __shared__ float sW[SWSZ];
  const int tid = threadIdx.x;
  const int wave = tid >> 5;
  const int lane = tid & 31;
  const int half = lane >> 4;
  const int lrow = lane & 15;
  const int m0 = blockIdx.x * 128 + wave * 16;
  int r = m0 + lrow;
  int rc = (r < rows) ? r : (rows - 1);

  v8f acc[8];
#pragma unroll
  for (int t = 0; t < 8; ++t) {
    const float bv = bias[t * 16 + lrow];  // each lane's column gets its bias
    v8f v;
#pragma unroll
    for (int i = 0; i < 8; ++i) v[i] = bv;
    acc[t] = v;
  }

  // ---- pass 1: x_self @ Wu_top ----
  pack_w(sW, Wu, tid);
  __syncthreads();
  {
    const float* arow = xs + (long)rc * D;
    for (int k0 = 0; k0 < D; k0 += 4) {
      const int kA = k0 + 2 * half;
      v2f a;
      a.x = arow[kA];
      a.y = arow[kA + 1];
      const int kp = (k0 >> 1) + half;
#pragma unroll
      for (int t = 0; t < 8; ++t) {
        const int n = t * 16 + lrow;
        const v2f b = *(const v2f*)&sW[(kp * PDIM + n) << 1];
        acc[t] = __builtin_amdgcn_wmma_f32_16x16x4_f32(
            false, a, false, b, (short)0, acc[t], false, false);
      }
    }
  }
  __syncthreads();

  // ---- pass 2: n_agg @ Wu_bot ----
  pack_w(sW, Wu + D * D, tid);
  __syncthreads();
  {
    float c = ncnt[rc];
    const float scale = 1.0f / (c > 1.0f ? c : 1.0f);
    const float* arow = nsum + (long)rc * D;
    for (int k0 = 0; k0 < D; k0 += 4) {
      const int kA = k0 + 2 * half;
      v2f a;
      a.x = arow[kA] * scale;
      a.y = arow[kA + 1] * scale;
      const int kp = (k0 >> 1) + half;
#pragma unroll
      for (int t = 0; t < 8; ++t) {
        const int n = t * 16 + lrow;
        const v2f b = *(const v2f*)&sW[(kp * PDIM + n) << 1];
        acc[t] = __builtin_amdgcn_wmma_f32_16x16x4_f32(
            false, a, false, b, (short)0, acc[t], false, false);
      }
    }
  }

#pragma unroll
  for (int t = 0; t < 8; ++t) {
    const int n = t * 16 + lrow;
#pragma unroll
    for (int rr = 0; rr < 8; ++rr) {
      const int m = m0 + rr + 8 * half;
      if (m < rows) {
        float v = acc[t][rr];
        out[(long)m * D + n] = v > 0.0f ? v : 0.0f;
      }
    }
  }
}

// ---------------------------------------------------------------------------
// Launch: zero scratch -> x@Wv -> scatter(v->e) -> e@We -> scatter(e->v) ->
//         fused concat-GEMM + bias + relu.
// ---------------------------------------------------------------------------
extern "C" void kernel_launch(void* const* d_in, const int* in_sizes, int n_in,
                              void* d_out, int out_size, void* d_ws,
                              size_t ws_size, hipStream_t stream) {
  const float* x  = (const float*)d_in[0];
  const int* edge = (const int*)d_in[1];   // [2, E] (JAX demotes int64->int32)
  const float* Wv = (const float*)d_in[2];
  const float* We = (const float*)d_in[3];
  const float* Wu = (const float*)d_in[4];
  const float* bu = (const float*)d_in[5];
  float* out = (float*)d_out;

  const int Nn = in_sizes[0] / D;   // 100000
  const int Ee = in_sizes[1] / 2;   // 600000
  const int Mm = 50000;             // hyperedges (fixed by reference)

  float* ws    = (float*)d_ws;
  float* xs    = ws;                          // N*128  (x @ W_v)
  float* esum  = xs + (long)Nn * D;           // M*128
  float* ecnt  = esum + (long)Mm * D;         // M
  float* eproj = ecnt + Mm;                   // M*128
  float* nsum  = eproj + (long)Mm * D;        // N*128
  float* ncnt  = nsum + (long)Nn * D;         // N

  const int* rowi = edge;        // node ids
  const int* coli = edge + Ee;   // hyperedge ids

  // zero everything from esum through ncnt (eproj zeroing is harmless)
  const long zn = ((long)Mm * D + Mm) + (long)Mm * D + ((long)Nn * D + Nn);
  hg_zero<<<1024, 256, 0, stream>>>(esum, zn);

  hg_gemm128<<<(Nn + 127) / 128, 256, 0, stream>>>(x, nullptr, Wv, xs, Nn);
  hg_scatter<<<(Ee + 7) / 8, 256, 0, stream>>>(xs, rowi, coli, esum, ecnt, Ee);
  hg_gemm128<<<(Mm + 127) / 128, 256, 0, stream>>>(esum, ecnt, We, eproj, Mm);
  hg_scatter<<<(Ee + 7) / 8, 256, 0, stream>>>(eproj, coli, rowi, nsum, ncnt, Ee);
  hg_final<<<(Nn + 127) / 128, 256, 0, stream>>>(xs, nsum, ncnt, Wu, bu, out, Nn);
}